// ForwardKinematics_59785944760784
// MI455X (gfx1250) — compile-verified
//
#include <hip/hip_runtime.h>

typedef float v2f __attribute__((ext_vector_type(2)));
typedef float v8f __attribute__((ext_vector_type(8)));

#define N_LINKS   (1 << 20)
#define BLOCK     256
#define ELEMS     4
#define GRID_A    (N_LINKS / (BLOCK * ELEMS))   /* 1024 blocks */
#define NAGG      GRID_A                        /* 1024 block aggregates */
#define BELEMS    (NAGG / BLOCK)                /* 4 aggregates/thread in scan kernel */

struct Aff { float r[9]; float p[3]; };   // affine: y = r*x + p   (r row-major 3x3)

__device__ __forceinline__ void setIdent(Aff &a) {
  #pragma unroll
  for (int i = 0; i < 9; ++i) a.r[i] = 0.f;
  a.r[0] = a.r[4] = a.r[8] = 1.f;
  a.p[0] = a.p[1] = a.p[2] = 0.f;
}

__device__ __forceinline__ void mat3mul(float *c, const float *a, const float *b) {
  #pragma unroll
  for (int i = 0; i < 3; ++i)
    #pragma unroll
    for (int j = 0; j < 3; ++j)
      c[3*i+j] = a[3*i+0]*b[0+j] + a[3*i+1]*b[3+j] + a[3*i+2]*b[6+j];
}

__device__ __forceinline__ void matvec(float *y, const float *a, const float *v) {
  #pragma unroll
  for (int i = 0; i < 3; ++i)
    y[i] = a[3*i+0]*v[0] + a[3*i+1]*v[1] + a[3*i+2]*v[2];
}

// a is the EARLIER prefix (left factor):  (a∘b)(x) = a(b(x))
__device__ __forceinline__ Aff compose(const Aff &a, const Aff &b) {
  Aff c;
  mat3mul(c.r, a.r, b.r);
  float t[3]; matvec(t, a.r, b.p);
  c.p[0] = a.p[0] + t[0]; c.p[1] = a.p[1] + t[1]; c.p[2] = a.p[2] + t[2];
  return c;
}

// R = Rz @ Ry @ Rx   (matches reference _euler_rots)
__device__ __forceinline__ void eulerR(float tx, float ty, float tz, float *R) {
  float sx, cx, sy, cy, sz, cz;
  __sincosf(tx, &sx, &cx); __sincosf(ty, &sy, &cy); __sincosf(tz, &sz, &cz);
  R[0] = cz*cy; R[1] = cz*sy*sx - sz*cx; R[2] = cz*sy*cx + sz*sx;
  R[3] = sz*cy; R[4] = sz*sy*sx + cz*cx; R[5] = sz*sy*cx - cz*sx;
  R[6] = -sy;   R[7] = cy*sx;            R[8] = cy*cx;
}

__device__ __forceinline__ Aff shflUpAff(const Aff &s, int d) {
  Aff o;
  #pragma unroll
  for (int k = 0; k < 9; ++k) o.r[k] = __shfl_up(s.r[k], d, 32);
  #pragma unroll
  for (int k = 0; k < 3; ++k) o.p[k] = __shfl_up(s.p[k], d, 32);
  return o;
}

// serial affine chain over this thread's ELEMS consecutive links
template <bool SAVE>
__device__ __forceinline__ void serialChain(const float *__restrict__ len,
                                            const float *__restrict__ theta,
                                            long base, Aff &S, float (*qser)[3]) {
  setIdent(S);
  #pragma unroll
  for (int e = 0; e < ELEMS; ++e) {
    long i = base + e;
    float R[9];
    eulerR(theta[3*i+0], theta[3*i+1], theta[3*i+2], R);
    float l = len[i];
    float t[3] = { l*R[2], l*R[5], l*R[8] };    // l * (third column of R)
    float tv[3]; matvec(tv, S.r, t);
    S.p[0] += tv[0]; S.p[1] += tv[1]; S.p[2] += tv[2];
    float nr[9]; mat3mul(nr, S.r, R);
    #pragma unroll
    for (int k = 0; k < 9; ++k) S.r[k] = nr[k];
    if constexpr (SAVE) { qser[e][0] = S.p[0]; qser[e][1] = S.p[1]; qser[e][2] = S.p[2]; }
  }
}

// -------------------------------------------------------------------------
// Exclusive cumsum of a 3-vector across the 32 lanes of a wave, in f32,
// via chained V_WMMA_F32_16X16X4_F32 with a triangular B matrix.
//   A rows 0..2  = components of elements  0..15  (sequence in K, chained)
//   A rows 8..10 = components of elements 16..31
//   B[p][n] = (p <= n)   -> C[m][n] = inclusive prefix at position n
// C layout: VGPR r, lanes 0-15 -> M=r (group0), lanes 16-31 -> M=r+8 (group1),
// so each lane's own local inclusive prefix lands in its own d[0..2].
// -------------------------------------------------------------------------
__device__ __forceinline__ void waveExclCumsum3_wmma(const float cv[3], int lane,
                                                     float qex[3]) {
  v8f acc = {0.f,0.f,0.f,0.f,0.f,0.f,0.f,0.f};
  const int r  = lane & 15;
  const int hi = lane >> 4;
  const bool actLo = (r < 3);
  const bool actHi = (r >= 8 && r < 11);
  const bool act   = actLo || actHi;
  const int comp   = actLo ? r : (actHi ? (r - 8) : 0);
  const int gbase  = actLo ? 0 : 16;          // which 16-element group this row band carries
  #pragma unroll
  for (int c = 0; c < 4; ++c) {
    // A layout (32-bit 16x4): lanes 0-15 hold K=0,1; lanes 16-31 hold K=2,3
    int e0 = gbase + 4*c + 2*hi;
    float g0a = __shfl(cv[0], e0,     32), g1a = __shfl(cv[1], e0,     32), g2a = __shfl(cv[2], e0,     32);
    float g0b = __shfl(cv[0], e0 + 1, 32), g1b = __shfl(cv[1], e0 + 1, 32), g2b = __shfl(cv[2], e0 + 1, 32);
    float av0 = (comp == 0) ? g0a : ((comp == 1) ? g1a : g2a);
    float av1 = (comp == 0) ? g0b : ((comp == 1) ? g1b : g2b);
    v2f a; a.x = act ? av0 : 0.f; a.y = act ? av1 : 0.f;
    int kb = 4*c + 2*hi;                       // chained sequence position of this lane's K
    v2f b; b.x = (kb     <= r) ? 1.f : 0.f;
           b.y = (kb + 1 <= r) ? 1.f : 0.f;
    acc = __builtin_amdgcn_wmma_f32_16x16x4_f32(false, a, false, b,
                                                (short)0, acc, false, false);
  }
  float inc[3] = { acc[0], acc[1], acc[2] };   // local inclusive prefix (own group)
  float ex[3], tot[3];
  #pragma unroll
  for (int k = 0; k < 3; ++k) {
    ex[k]  = __shfl_up(inc[k], 1, 32);         // lane16 picks up group0 total from lane15
    tot[k] = __shfl(inc[k], 15, 32);           // group0 total
  }
  if (lane == 0)  { ex[0] = ex[1] = ex[2] = 0.f; }
  if (lane >= 17) { ex[0] += tot[0]; ex[1] += tot[1]; ex[2] += tot[2]; }
  qex[0] = ex[0]; qex[1] = ex[1]; qex[2] = ex[2];
}

// =========================================================================
// Kernel A: per-block affine aggregate
// =========================================================================
__global__ __launch_bounds__(BLOCK)
void fk_partials(const float *__restrict__ len, const float *__restrict__ theta,
                 float *__restrict__ agg) {
  const int tid = threadIdx.x, lane = tid & 31, wave = tid >> 5;
  const long base = ((long)blockIdx.x * BLOCK + tid) * ELEMS;
  Aff S;
  serialChain<false>(len, theta, base, S, nullptr);
  // wave-level inclusive affine scan (Kogge-Stone)
  #pragma unroll
  for (int d = 1; d < 32; d <<= 1) {
    Aff o = shflUpAff(S, d);
    if (lane >= d) S = compose(o, S);
  }
  __shared__ float sm[8][12];
  if (lane == 31) {
    #pragma unroll
    for (int k = 0; k < 9; ++k) sm[wave][k] = S.r[k];
    sm[wave][9] = S.p[0]; sm[wave][10] = S.p[1]; sm[wave][11] = S.p[2];
  }
  __syncthreads();
  if (tid == 0) {
    Aff B;
    #pragma unroll
    for (int k = 0; k < 9; ++k) B.r[k] = sm[0][k];
    B.p[0] = sm[0][9]; B.p[1] = sm[0][10]; B.p[2] = sm[0][11];
    #pragma unroll
    for (int w = 1; w < 8; ++w) {
      Aff W;
      #pragma unroll
      for (int k = 0; k < 9; ++k) W.r[k] = sm[w][k];
      W.p[0] = sm[w][9]; W.p[1] = sm[w][10]; W.p[2] = sm[w][11];
      B = compose(B, W);
    }
    float *g = agg + (long)blockIdx.x * 12;
    #pragma unroll
    for (int k = 0; k < 9; ++k) g[k] = B.r[k];
    g[9] = B.p[0]; g[10] = B.p[1]; g[11] = B.p[2];
  }
}

// =========================================================================
// Kernel B: exclusive scan of the 1024 block aggregates (one block)
// =========================================================================
__global__ __launch_bounds__(BLOCK)
void fk_blockscan(const float *__restrict__ agg, float *__restrict__ pre) {
  const int tid = threadIdx.x, lane = tid & 31, wave = tid >> 5;
  Aff ser[BELEMS];
  Aff S; setIdent(S);
  #pragma unroll
  for (int e = 0; e < BELEMS; ++e) {
    const float *g = agg + (long)(tid * BELEMS + e) * 12;
    Aff A;
    #pragma unroll
    for (int k = 0; k < 9; ++k) A.r[k] = g[k];
    A.p[0] = g[9]; A.p[1] = g[10]; A.p[2] = g[11];
    S = compose(S, A);
    ser[e] = S;
  }
  Aff T = S;
  #pragma unroll
  for (int d = 1; d < 32; d <<= 1) {
    Aff o = shflUpAff(T, d);
    if (lane >= d) T = compose(o, T);
  }
  Aff lex = shflUpAff(T, 1);
  if (lane == 0) setIdent(lex);
  __shared__ float sm[8][12];
  if (lane == 31) {
    #pragma unroll
    for (int k = 0; k < 9; ++k) sm[wave][k] = T.r[k];
    sm[wave][9] = T.p[0]; sm[wave][10] = T.p[1]; sm[wave][11] = T.p[2];
  }
  __syncthreads();
  Aff wp; setIdent(wp);
  for (int w = 0; w < wave; ++w) {
    Aff W;
    #pragma unroll
    for (int k = 0; k < 9; ++k) W.r[k] = sm[w][k];
    W.p[0] = sm[w][9]; W.p[1] = sm[w][10]; W.p[2] = sm[w][11];
    wp = compose(wp, W);
  }
  Aff tex = compose(wp, lex);
  #pragma unroll
  for (int e = 0; e < BELEMS; ++e) {
    Aff P = (e == 0) ? tex : compose(tex, ser[e - 1]);
    float *g = pre + (long)(tid * BELEMS + e) * 12;
    #pragma unroll
    for (int k = 0; k < 9; ++k) g[k] = P.r[k];
    g[9] = P.p[0]; g[10] = P.p[1]; g[11] = P.p[2];
  }
}

// =========================================================================
// Kernel C: recompute local chains, full scan, write positions
// =========================================================================
__global__ __launch_bounds__(BLOCK)
void fk_apply(const float *__restrict__ len, const float *__restrict__ theta,
              const float *__restrict__ pre, float *__restrict__ out) {
  const int tid = threadIdx.x, lane = tid & 31, wave = tid >> 5;
  const long base = ((long)blockIdx.x * BLOCK + tid) * ELEMS;
  Aff S; float qser[ELEMS][3];
  serialChain<true>(len, theta, base, S, qser);

  // rotation-only inclusive wave scan (non-commutative -> shuffle Kogge-Stone)
  float Ri[9];
  #pragma unroll
  for (int k = 0; k < 9; ++k) Ri[k] = S.r[k];
  #pragma unroll
  for (int d = 1; d < 32; d <<= 1) {
    float o[9];
    #pragma unroll
    for (int k = 0; k < 9; ++k) o[k] = __shfl_up(Ri[k], d, 32);
    if (lane >= d) {
      float t[9]; mat3mul(t, o, Ri);
      #pragma unroll
      for (int k = 0; k < 9; ++k) Ri[k] = t[k];
    }
  }
  float Rex[9];
  #pragma unroll
  for (int k = 0; k < 9; ++k) Rex[k] = __shfl_up(Ri[k], 1, 32);
  if (lane == 0) {
    #pragma unroll
    for (int k = 0; k < 9; ++k) Rex[k] = 0.f;
    Rex[0] = Rex[4] = Rex[8] = 1.f;
  }

  // additive part across lanes: f32 WMMA triangular cumsum (EXEC all ones here)
  float cv[3]; matvec(cv, Rex, S.p);
  float qex[3]; waveExclCumsum3_wmma(cv, lane, qex);

  __shared__ float sm[8][12];
  if (lane == 31) {
    #pragma unroll
    for (int k = 0; k < 9; ++k) sm[wave][k] = Ri[k];
    sm[wave][9]  = qex[0] + cv[0];
    sm[wave][10] = qex[1] + cv[1];
    sm[wave][11] = qex[2] + cv[2];
  }
  __syncthreads();

  Aff TP;
  {
    const float *bp = pre + (long)blockIdx.x * 12;
    #pragma unroll
    for (int k = 0; k < 9; ++k) TP.r[k] = bp[k];
    TP.p[0] = bp[9]; TP.p[1] = bp[10]; TP.p[2] = bp[11];
  }
  for (int w = 0; w < wave; ++w) {
    Aff W;
    #pragma unroll
    for (int k = 0; k < 9; ++k) W.r[k] = sm[w][k];
    W.p[0] = sm[w][9]; W.p[1] = sm[w][10]; W.p[2] = sm[w][11];
    TP = compose(TP, W);
  }
  Aff L;
  #pragma unroll
  for (int k = 0; k < 9; ++k) L.r[k] = Rex[k];
  L.p[0] = qex[0]; L.p[1] = qex[1]; L.p[2] = qex[2];
  TP = compose(TP, L);

  #pragma unroll
  for (int e = 0; e < ELEMS; ++e) {
    float o3[3]; matvec(o3, TP.r, qser[e]);
    long i = base + e;
    out[3*i+0] = TP.p[0] + o3[0];
    out[3*i+1] = TP.p[1] + o3[1];
    out[3*i+2] = TP.p[2] + o3[2];
  }
}

extern "C" void kernel_launch(void* const* d_in, const int* in_sizes, int n_in,
                              void* d_out, int out_size, void* d_ws, size_t ws_size,
                              hipStream_t stream) {
  (void)in_sizes; (void)n_in; (void)out_size; (void)ws_size;
  const float* len   = (const float*)d_in[0];   // [N]
  const float* theta = (const float*)d_in[1];   // [N,3]
  float* out = (float*)d_out;                   // [N,3]
  float* ws  = (float*)d_ws;
  float* agg = ws;                              // NAGG * 12 floats
  float* pre = ws + (size_t)NAGG * 12;          // NAGG * 12 floats  (96 KB total)
  fk_partials <<<GRID_A, BLOCK, 0, stream>>>(len, theta, agg);
  fk_blockscan<<<1,      BLOCK, 0, stream>>>(agg, pre);
  fk_apply    <<<GRID_A, BLOCK, 0, stream>>>(len, theta, pre, out);
}